// GetFilters_15796889715106
// MI455X (gfx1250) — compile-verified
//
#include <hip/hip_runtime.h>
#include <hip/hip_bf16.h>
#include <math.h>

// CDNA5 / gfx1250: wave32, f32 WMMA (16x16x4) + async global->LDS staging

typedef __attribute__((ext_vector_type(2))) float v2f;
typedef __attribute__((ext_vector_type(8))) float v8f;

namespace {
constexpr int kN      = 4;
constexpr int kC      = 64;
constexpr int kH      = 384;
constexpr int kW      = 768;
constexpr int kR      = 2;          // RADIUS
constexpr int kWin    = 25;         // (2R+1)^2
constexpr int kRowsWG = 8;          // one output row per wave, 8 waves
constexpr int kTW     = 12;         // useful output columns per 16-wide WMMA tile
constexpr int kLdsRows = kRowsWG + 2 * kR;  // 12 rows incl. vertical halo
constexpr int kCStr   = kC + 1;     // 65: pad channel stride to avoid LDS bank conflicts
constexpr float kEps  = 1e-12f;
}

__global__ __launch_bounds__(256)
void GetFilters_15796889715106_kernel(const float* __restrict__ x,
                                      float* __restrict__ out) {
    // Tile of normalized pixels: [row][col][channel], channel-padded stride
    __shared__ float xs[kLdsRows][16][kCStr];           // 49,920 B
    // Per-wave diagonal scratch: [wave][dr][m][dc]
    __shared__ float diag[kRowsWG][5][kTW][5];          //  9,600 B

    const int tid = threadIdx.x;
    const int n   = blockIdx.z;
    const int h0  = blockIdx.y * kRowsWG;
    const int w0  = blockIdx.x * kTW;
    const int wb  = w0 - kR;        // global w of LDS column 0

    // Token CDNA5 prefetch of this tile's first cacheline (global_prefetch_b8)
    __builtin_prefetch(x + (((size_t)n * kC) * kH + h0) * kW + w0, 0, 1);

    // ---- border blocks: pre-zero the tile so skipped (out-of-range) async
    //      lanes leave reference-matching zero padding behind ----
    const bool border = (blockIdx.x == 0) || (blockIdx.x == gridDim.x - 1) ||
                        (blockIdx.y == 0) || (blockIdx.y == gridDim.y - 1);
    if (border) {
        float* flat = &xs[0][0][0];
        for (int e = tid; e < kLdsRows * 16 * kCStr; e += 256)
            flat[e] = 0.0f;
    }
    __syncthreads();   // DS zero-stores complete before async writes land

    // ---- async global->LDS staging: 12 rows x 16 cols x 64 ch ----
    // e = (c*12 + r)*16 + col : 16 consecutive lanes hit 16 consecutive w (64B)
    for (int e = tid; e < kLdsRows * 16 * kC; e += 256) {
        int col = e & 15;
        int r   = (e >> 4) % kLdsRows;
        int c   = e / (16 * kLdsRows);
        int h   = h0 - kR + r;
        int w   = wb + col;
        if (h >= 0 && h < kH && w >= 0 && w < kW) {
            // low 32 bits of a generic LDS pointer == wave-relative LDS byte addr
            unsigned lb = (unsigned)(size_t)&xs[r][col][c];
            const float* gp = &x[(((size_t)n * kC + c) * kH + h) * kW + w];
            asm volatile("global_load_async_to_lds_b32 %0, %1, off"
                         :: "v"(lb), "v"(gp) : "memory");
        }
    }
    asm volatile("s_wait_asynccnt 0" ::: "memory");  // each wave drains its ASYNCcnt
    __syncthreads();

    // ---- in-place L2 normalization over channels (192 pixels, 1 per thread) ----
    if (tid < kLdsRows * 16) {
        int r = tid >> 4, col = tid & 15;
        float s = 0.0f;
        #pragma unroll
        for (int c = 0; c < kC; ++c) { float v = xs[r][col][c]; s += v * v; }
        float inv = 1.0f / fmaxf(sqrtf(s), kEps);
        #pragma unroll
        for (int c = 0; c < kC; ++c) xs[r][col][c] *= inv;
    }
    __syncthreads();

    // ---- WMMA correlation: D[m][nn] = dot(pix(h, wb+m), pix(h+dr-2, wb+nn)) ----
    const int wv   = tid >> 5;      // wave id = output row within tile
    const int lane = tid & 31;
    const int mn   = lane & 15;     // A row m / B col n
    const int half = lane >> 4;     // K-pair selector (f32 16x16x4 layout)
    const int rA   = wv + kR;       // LDS row of this wave's output row

    // Preload A operand (reused across all 5 dr): K = kc*4 + 2*half + {0,1}
    v2f a[16];
    #pragma unroll
    for (int kc = 0; kc < 16; ++kc) {
        int kb = kc * 4 + 2 * half;
        v2f t;
        t.x = xs[rA][mn][kb];
        t.y = xs[rA][mn][kb + 1];
        a[kc] = t;
    }

    for (int dr = 0; dr < 5; ++dr) {
        int rB = wv + dr;           // rows h0+wv-2 .. h0+wv+2 (in-tile, halo covered)
        v8f acc = {};
        #pragma unroll
        for (int kc = 0; kc < 16; ++kc) {
            int kb = kc * 4 + 2 * half;
            v2f b;
            b.x = xs[rB][mn][kb];
            b.y = xs[rB][mn][kb + 1];
            acc = __builtin_amdgcn_wmma_f32_16x16x4_f32(
                false, a[kc], false, b, (short)0, acc, false, false);
        }
        // D element (M,N) lives in VGPR M%8, lane N + 16*(M/8).
        // Keep diagonals |N-M|<=2 for interior rows M in [2,13].
        #pragma unroll
        for (int vg = 0; vg < 8; ++vg) {
            int M = vg + 8 * half;
            int d = mn - M;
            if (M >= kR && M < 16 - kR && d >= -kR && d <= kR)
                diag[wv][dr][M - kR][d + kR] = acc[vg];
        }
    }

    // ---- L1 normalize over the 25 window values and store (same-wave DS order) ----
    if (lane < kTW) {
        float s = 0.0f;
        #pragma unroll
        for (int k = 0; k < kWin; ++k)
            s += fabsf(diag[wv][k / 5][lane][k % 5]);
        float inv = 1.0f / fmaxf(s, kEps);
        int hrow = h0 + wv;
        int w    = w0 + lane;
        #pragma unroll
        for (int k = 0; k < kWin; ++k)
            out[(((size_t)n * kWin + k) * kH + hrow) * kW + w] =
                diag[wv][k / 5][lane][k % 5] * inv;
    }
}

extern "C" void kernel_launch(void* const* d_in, const int* in_sizes, int n_in,
                              void* d_out, int out_size, void* d_ws, size_t ws_size,
                              hipStream_t stream) {
    const float* x = (const float*)d_in[0];
    float* out     = (float*)d_out;
    (void)in_sizes; (void)n_in; (void)out_size; (void)d_ws; (void)ws_size;

    dim3 grid(kW / kTW, kH / kRowsWG, kN);   // (64, 48, 4)
    GetFilters_15796889715106_kernel<<<grid, 256, 0, stream>>>(x, out);
}